// MultiHeadSGATLayer_63093069578621
// MI455X (gfx1250) — compile-verified
//
#include <hip/hip_runtime.h>
#include <math.h>

typedef __attribute__((ext_vector_type(2))) float v2f;
typedef __attribute__((ext_vector_type(4))) float v4f;
typedef __attribute__((ext_vector_type(8))) float v8f;

#define IN_DIM   128
#define OUT_DIM  32
#define N_HEADS  8
#define FEATS    (N_HEADS * OUT_DIM)   // 256
#define NEG_SLOPE 0.01f
#define NODES_PER_BLK 4                // aggregation: 4 nodes x 64 threads

// ---------------------------------------------------------------------------
// Kernel 1: z[N,256] = h[N,128] @ Wcat[128,256] using V_WMMA_F32_16X16X4_F32.
// One wave per 16x16 tile, K-loop of 32 wmma steps. A column tile of 16 lies
// entirely inside one head, so the B panel is W[head][k][o0 + lane].
// A layout (ISA 7.12.2, 32-bit A 16x4): lanes 0-15 hold K=k,k+1 in v[0],v[1];
// lanes 16-31 hold K=k+2,k+3. B mirrored (row of B striped across lanes).
// ---------------------------------------------------------------------------
__global__ __launch_bounds__(256)
void gat_gemm_wmma(const float* __restrict__ h, const float* __restrict__ W,
                   float* __restrict__ z, int n_nodes)
{
    const int wave = threadIdx.x >> 5;          // 0..7
    const int lane = threadIdx.x & 31;
    const int half = lane >> 4;                 // 0 or 1
    const int l    = lane & 15;
    const int row0 = blockIdx.x * 16;
    if (row0 >= n_nodes) return;
    const int tile = blockIdx.y * 8 + wave;     // 0..15 column tile
    const int col0 = tile * 16;
    const int head = tile >> 1;
    const int o0   = (tile & 1) * 16;

    const float* __restrict__ Arow = h + (size_t)(row0 + l) * IN_DIM + 2 * half;
    const float* __restrict__ Wh   = W + (size_t)head * (IN_DIM * OUT_DIM) + o0 + l;

    v8f c = {};
    #pragma unroll 4
    for (int k = 0; k < IN_DIM; k += 4) {
        v2f a = *(const v2f*)(Arow + k);        // A[M=l][K=k+2*half+{0,1}]
        const int kb = k + 2 * half;
        v2f b;                                  // B[K=kb+{0,1}][N=col0+l]
        b.x = Wh[(kb + 0) * OUT_DIM];
        b.y = Wh[(kb + 1) * OUT_DIM];
        c = __builtin_amdgcn_wmma_f32_16x16x4_f32(
                false, a, false, b, (short)0, c, false, false);
    }

    // D layout: VGPR r, lanes 0-15 -> (M=r, N=l); lanes 16-31 -> (M=r+8, N=l)
    float* __restrict__ zo = z + col0 + l;
    #pragma unroll
    for (int r = 0; r < 8; ++r)
        zo[(size_t)(row0 + r + 8 * half) * FEATS] = c[r];
}

// ---------------------------------------------------------------------------
// Kernel 2: el[n][h] = z[n,h,:] . a_src[h,:] ; er likewise with a_dst.
// ---------------------------------------------------------------------------
__global__ __launch_bounds__(256)
void gat_attn_coeff(const float* __restrict__ z,
                    const float* __restrict__ a_src,
                    const float* __restrict__ a_dst,
                    float* __restrict__ el, float* __restrict__ er,
                    int n_nodes)
{
    int t = blockIdx.x * blockDim.x + threadIdx.x;
    if (t >= n_nodes * N_HEADS) return;
    const int n  = t >> 3;
    const int hd = t & 7;
    const v4f* __restrict__ zr = (const v4f*)(z + (size_t)n * FEATS + hd * OUT_DIM);
    const v4f* __restrict__ as = (const v4f*)(a_src + hd * OUT_DIM);
    const v4f* __restrict__ ad = (const v4f*)(a_dst + hd * OUT_DIM);
    float sl = 0.f, sr = 0.f;
    #pragma unroll
    for (int q = 0; q < OUT_DIM / 4; ++q) {
        v4f v = zr[q], s = as[q], d = ad[q];
        sl = fmaf(v.x, s.x, fmaf(v.y, s.y, fmaf(v.z, s.z, fmaf(v.w, s.w, sl))));
        sr = fmaf(v.x, d.x, fmaf(v.y, d.y, fmaf(v.z, d.z, fmaf(v.w, d.w, sr))));
    }
    el[t] = sl;
    er[t] = sr;
}

// ---------------------------------------------------------------------------
// Kernel 3: dst is sorted -> CSR row offsets via binary search.
// ---------------------------------------------------------------------------
__global__ __launch_bounds__(256)
void gat_row_start(const int* __restrict__ dst, int n_edges, int n_nodes,
                   int* __restrict__ row_start)
{
    int n = blockIdx.x * blockDim.x + threadIdx.x;
    if (n > n_nodes) return;
    int lo = 0, hi = n_edges;
    while (lo < hi) {
        int mid = (lo + hi) >> 1;
        if (dst[mid] < n) lo = mid + 1; else hi = mid;
    }
    row_start[n] = lo;
}

// ---------------------------------------------------------------------------
// Kernel 4: per-destination-node fused softmax + aggregation.
// 256-thread block = 4 node-groups x 64 threads. Thread owns 4 consecutive
// features (b128 loads/stores). Pass1: per-head max; pass2: per-head sum-exp
// (per-group LDS tree reduce); pass3: loop node's edges, acc4 += alpha *
// z[src][f4] via global_load_b128. No atomics; deterministic. Empty segments
// fall through (max-guard -> 0, denom -> 1e-9, acc = 0) so barriers stay
// uniform across the block.
// ---------------------------------------------------------------------------
__global__ __launch_bounds__(256)
void gat_aggregate(const float* __restrict__ z,
                   const float* __restrict__ el,
                   const float* __restrict__ er,
                   const int* __restrict__ src,
                   const int* __restrict__ row_start,
                   float* __restrict__ out, int n_nodes)
{
    __shared__ float sred[NODES_PER_BLK][N_HEADS][64];
    __shared__ float smax[NODES_PER_BLK][N_HEADS];
    __shared__ float sinv[NODES_PER_BLK][N_HEADS];

    const int tid = threadIdx.x;
    const int g   = tid >> 6;              // node group 0..3
    const int gt  = tid & 63;              // thread within group
    const int n   = blockIdx.x * NODES_PER_BLK + g;
    const bool live = (n < n_nodes);

    int start = 0, end = 0;
    if (live) { start = row_start[n]; end = row_start[n + 1]; }

    float ern_r[N_HEADS];
    #pragma unroll
    for (int hh = 0; hh < N_HEADS; ++hh) ern_r[hh] = 0.f;
    if (live) {
        const v4f e0 = *(const v4f*)(er + (size_t)n * N_HEADS);
        const v4f e1 = *(const v4f*)(er + (size_t)n * N_HEADS + 4);
        ern_r[0] = e0.x; ern_r[1] = e0.y; ern_r[2] = e0.z; ern_r[3] = e0.w;
        ern_r[4] = e1.x; ern_r[5] = e1.y; ern_r[6] = e1.z; ern_r[7] = e1.w;
    }

    // ---- pass 1: per-head max over this node's edges ----
    float acc8[N_HEADS];
    #pragma unroll
    for (int hh = 0; hh < N_HEADS; ++hh) acc8[hh] = -3.4e38f;
    for (int j = start + gt; j < end; j += 64) {
        const int s = src[j];
        const v4f l0 = *(const v4f*)(el + (size_t)s * N_HEADS);
        const v4f l1 = *(const v4f*)(el + (size_t)s * N_HEADS + 4);
        const float els[N_HEADS] = { l0.x, l0.y, l0.z, l0.w,
                                     l1.x, l1.y, l1.z, l1.w };
        #pragma unroll
        for (int hh = 0; hh < N_HEADS; ++hh) {
            float e = els[hh] + ern_r[hh];
            e = (e > 0.f) ? e : NEG_SLOPE * e;
            acc8[hh] = fmaxf(acc8[hh], e);
        }
    }
    #pragma unroll
    for (int hh = 0; hh < N_HEADS; ++hh) sred[g][hh][gt] = acc8[hh];
    __syncthreads();
    for (int off = 32; off > 0; off >>= 1) {
        if (gt < off) {
            #pragma unroll
            for (int hh = 0; hh < N_HEADS; ++hh)
                sred[g][hh][gt] = fmaxf(sred[g][hh][gt], sred[g][hh][gt + off]);
        }
        __syncthreads();
    }
    if (gt < N_HEADS) {
        float m = sred[g][gt][0];
        smax[g][gt] = (m > -1e38f) ? m : 0.f;    // -inf guard like reference
    }
    __syncthreads();
    float mx[N_HEADS];
    #pragma unroll
    for (int hh = 0; hh < N_HEADS; ++hh) mx[hh] = smax[g][hh];
    __syncthreads();

    // ---- pass 2: per-head sum of exp(e - max) ----
    #pragma unroll
    for (int hh = 0; hh < N_HEADS; ++hh) acc8[hh] = 0.f;
    for (int j = start + gt; j < end; j += 64) {
        const int s = src[j];
        const v4f l0 = *(const v4f*)(el + (size_t)s * N_HEADS);
        const v4f l1 = *(const v4f*)(el + (size_t)s * N_HEADS + 4);
        const float els[N_HEADS] = { l0.x, l0.y, l0.z, l0.w,
                                     l1.x, l1.y, l1.z, l1.w };
        #pragma unroll
        for (int hh = 0; hh < N_HEADS; ++hh) {
            float e = els[hh] + ern_r[hh];
            e = (e > 0.f) ? e : NEG_SLOPE * e;
            acc8[hh] += __expf(e - mx[hh]);
        }
    }
    #pragma unroll
    for (int hh = 0; hh < N_HEADS; ++hh) sred[g][hh][gt] = acc8[hh];
    __syncthreads();
    for (int off = 32; off > 0; off >>= 1) {
        if (gt < off) {
            #pragma unroll
            for (int hh = 0; hh < N_HEADS; ++hh)
                sred[g][hh][gt] += sred[g][hh][gt + off];
        }
        __syncthreads();
    }
    if (gt < N_HEADS)
        sinv[g][gt] = 1.f / fmaxf(sred[g][gt][0], 1e-9f);
    __syncthreads();

    // ---- pass 3: feature-parallel aggregation (float4 per thread) ----
    const int f4    = gt * 4;                // features f4..f4+3, same head
    const int head  = f4 >> 5;
    const float m_h  = smax[g][head];
    const float iv_h = sinv[g][head];
    const float er_h = ern_r[head];
    v4f acc = {0.f, 0.f, 0.f, 0.f};
    for (int j = start; j < end; ++j) {      // j uniform within group
        const int s = src[j];
        float e = el[(size_t)s * N_HEADS + head] + er_h;
        e = (e > 0.f) ? e : NEG_SLOPE * e;
        const float alpha = __expf(e - m_h) * iv_h;
        const v4f zv = *(const v4f*)(z + (size_t)s * FEATS + f4);
        acc.x = fmaf(alpha, zv.x, acc.x);
        acc.y = fmaf(alpha, zv.y, acc.y);
        acc.z = fmaf(alpha, zv.z, acc.z);
        acc.w = fmaf(alpha, zv.w, acc.w);
    }
    if (live)
        *(v4f*)(out + (size_t)n * FEATS + f4) = acc;
}

// ---------------------------------------------------------------------------
extern "C" void kernel_launch(void* const* d_in, const int* in_sizes, int n_in,
                              void* d_out, int out_size, void* d_ws, size_t ws_size,
                              hipStream_t stream)
{
    const float* h     = (const float*)d_in[0];
    const int*   src   = (const int*)d_in[1];
    const int*   dst   = (const int*)d_in[2];
    const float* W     = (const float*)d_in[3];
    const float* a_src = (const float*)d_in[4];
    const float* a_dst = (const float*)d_in[5];
    float* out = (float*)d_out;

    const int n_nodes = in_sizes[0] / IN_DIM;   // 50000
    const int n_edges = in_sizes[1];            // 800000

    // workspace layout
    float* z  = (float*)d_ws;                              // n_nodes*256
    float* el = z  + (size_t)n_nodes * FEATS;              // n_nodes*8
    float* er = el + (size_t)n_nodes * N_HEADS;            // n_nodes*8
    int*   rs = (int*)(er + (size_t)n_nodes * N_HEADS);    // n_nodes+1

    // 1) WMMA GEMM: grid (N/16, 2), 8 waves/block -> 16 col tiles
    {
        dim3 grid((n_nodes + 15) / 16, 2);
        gat_gemm_wmma<<<grid, 256, 0, stream>>>(h, W, z, n_nodes);
    }
    // 2) el / er
    {
        int total = n_nodes * N_HEADS;
        gat_attn_coeff<<<(total + 255) / 256, 256, 0, stream>>>(
            z, a_src, a_dst, el, er, n_nodes);
    }
    // 3) CSR offsets from sorted dst
    gat_row_start<<<(n_nodes + 1 + 255) / 256, 256, 0, stream>>>(
        dst, n_edges, n_nodes, rs);
    // 4) fused softmax + aggregation, 4 nodes per 256-thread block
    gat_aggregate<<<(n_nodes + NODES_PER_BLK - 1) / NODES_PER_BLK, 256, 0,
                    stream>>>(z, el, er, src, rs, out, n_nodes);
}